// OptimalTransportLoss_10428180594920
// MI455X (gfx1250) — compile-verified
//
#include <hip/hip_runtime.h>
#include <math.h>

// ---------------- problem constants (fixed by reference setup) ----------------
#define NPTS   4096        // B: rows of X and Y
#define DDIM   512         // feature dim
#define INV_REG 20.0f      // 1/0.05
#define REGF    0.05f
#define MARG    (1.0f/4096.0f)
#define EPSF    1e-9f
#define CAPF    1000.0f
#define ITERS   50
#define LOSS_BLOCKS 2048

typedef float v2f __attribute__((ext_vector_type(2)));
typedef float v8f __attribute__((ext_vector_type(8)));

// ---------------- init u = v = 1 ----------------
__global__ void init_uv(float* __restrict__ u, float* __restrict__ v) {
    int i = blockIdx.x * blockDim.x + threadIdx.x;
    if (i < NPTS) { u[i] = 1.0f; v[i] = 1.0f; }
}

// ---------------- per-row squared norms of X and Y ----------------
__global__ void row_norms(const float* __restrict__ X, const float* __restrict__ Y,
                          float* __restrict__ Xs, float* __restrict__ Ys) {
    __shared__ float sx[256];
    __shared__ float sy[256];
    const int r = blockIdx.x, t = threadIdx.x;
    const float* xr = X + (size_t)r * DDIM;
    const float* yr = Y + (size_t)r * DDIM;
    float ax = 0.0f, ay = 0.0f;
    for (int c = t; c < DDIM; c += 256) {
        float x = xr[c]; ax += x * x;
        float y = yr[c]; ay += y * y;
    }
    sx[t] = ax; sy[t] = ay;
    __syncthreads();
    for (int s = 128; s > 0; s >>= 1) {
        if (t < s) { sx[t] += sx[t + s]; sy[t] += sy[t + s]; }
        __syncthreads();
    }
    if (t == 0) { Xs[r] = sx[0]; Ys[r] = sy[0]; }
}

// ---------------- fused GEMM (X @ Y^T) + cost + K = exp(-cost/reg) ----------------
// Block: 256 threads = 8 waves, block tile 64(M) x 128(N).
// Wave tile: 32x32 = 2x2 WMMA f32 16x16x4 accumulators. K-chunk = 32 staged in LDS.
// LDS row stride 35 floats: 35 is coprime with 64 banks and cross-half-wave
// offset (+2 floats) maps to lane-delta 22 (>15) => conflict-free fragment reads.
__global__ __launch_bounds__(256) void gemm_expk(
        const float* __restrict__ X, const float* __restrict__ Y,
        const float* __restrict__ Xs, const float* __restrict__ Ys,
        float* __restrict__ Km) {
    __shared__ float As[64][35];
    __shared__ float Bs[128][35];

    const int t    = threadIdx.x;
    const int lane = t & 31;
    const int wid  = t >> 5;
    const int l15  = lane & 15;
    const int khi  = lane >> 4;         // 0: K={0,1}, 1: K={2,3} within a k4 step
    const int wm   = wid >> 2;          // 0..1
    const int wn   = wid & 3;           // 0..3
    const int m0   = blockIdx.y * 64;
    const int n0   = blockIdx.x * 128;

    v8f acc[2][2] = {};                 // zero-initialized accumulators

    for (int kt = 0; kt < DDIM; kt += 32) {
        // Cooperative load A tile (64 x 32): 512 float4 slots, 2 per thread.
        #pragma unroll
        for (int i = 0; i < 2; ++i) {
            int slot = t + 256 * i;
            int row = slot >> 3, c4 = (slot & 7) * 4;
            float4 val = *(const float4*)(X + (size_t)(m0 + row) * DDIM + kt + c4);
            As[row][c4 + 0] = val.x; As[row][c4 + 1] = val.y;
            As[row][c4 + 2] = val.z; As[row][c4 + 3] = val.w;
        }
        // Cooperative load B tile (128 x 32): 1024 float4 slots, 4 per thread.
        #pragma unroll
        for (int i = 0; i < 4; ++i) {
            int slot = t + 256 * i;
            int row = slot >> 3, c4 = (slot & 7) * 4;
            float4 val = *(const float4*)(Y + (size_t)(n0 + row) * DDIM + kt + c4);
            Bs[row][c4 + 0] = val.x; Bs[row][c4 + 1] = val.y;
            Bs[row][c4 + 2] = val.z; Bs[row][c4 + 3] = val.w;
        }
        __syncthreads();

        #pragma unroll
        for (int kk = 0; kk < 32; kk += 4) {
            const int kc = kk + khi * 2;
            // A fragments (16x4 f32): lane l15 -> row, 2 consecutive k per lane.
            v2f a0 = { As[wm * 32 +      l15][kc], As[wm * 32 +      l15][kc + 1] };
            v2f a1 = { As[wm * 32 + 16 + l15][kc], As[wm * 32 + 16 + l15][kc + 1] };
            // B fragments (4x16 f32): lane l15 -> column n, 2 consecutive k per lane.
            v2f b0 = { Bs[wn * 32 +      l15][kc], Bs[wn * 32 +      l15][kc + 1] };
            v2f b1 = { Bs[wn * 32 + 16 + l15][kc], Bs[wn * 32 + 16 + l15][kc + 1] };

            acc[0][0] = __builtin_amdgcn_wmma_f32_16x16x4_f32(false, a0, false, b0,
                          (short)0, acc[0][0], false, false);
            acc[0][1] = __builtin_amdgcn_wmma_f32_16x16x4_f32(false, a0, false, b1,
                          (short)0, acc[0][1], false, false);
            acc[1][0] = __builtin_amdgcn_wmma_f32_16x16x4_f32(false, a1, false, b0,
                          (short)0, acc[1][0], false, false);
            acc[1][1] = __builtin_amdgcn_wmma_f32_16x16x4_f32(false, a1, false, b1,
                          (short)0, acc[1][1], false, false);
        }
        __syncthreads();
    }

    // Epilogue: C/D layout -> VGPR r holds (M = base + r + 8*khi, N = base + l15).
    #pragma unroll
    for (int mi = 0; mi < 2; ++mi) {
        #pragma unroll
        for (int ni = 0; ni < 2; ++ni) {
            const int gn = n0 + wn * 32 + ni * 16 + l15;
            const float ysv = Ys[gn];
            #pragma unroll
            for (int r = 0; r < 8; ++r) {
                const int gm = m0 + wm * 32 + mi * 16 + khi * 8 + r;
                float cost = Xs[gm] + ysv - 2.0f * acc[mi][ni][r];
                cost = fminf(fmaxf(cost, 0.0f), CAPF);
                Km[(size_t)gm * NPTS + gn] = expf(-cost * INV_REG);
            }
        }
    }
}

// ---------------- u = a / (K @ v + eps) : one block per row ----------------
__global__ void matvec_row_u(const float* __restrict__ Km, const float* __restrict__ v,
                             float* __restrict__ u) {
    __shared__ float s[256];
    const int row = blockIdx.x, t = threadIdx.x;
    const float* kr = Km + (size_t)row * NPTS;
    float acc = 0.0f;
    #pragma unroll 4
    for (int c = t; c < NPTS; c += 256) acc += kr[c] * v[c];
    s[t] = acc;
    __syncthreads();
    for (int st = 128; st > 0; st >>= 1) {
        if (t < st) s[t] += s[t + st];
        __syncthreads();
    }
    if (t == 0) u[row] = MARG / (s[0] + EPSF);
}

// ---------------- partial K^T @ u : grid (16 col-groups, 32 row-chunks) ----------------
__global__ void matvec_col_partial(const float* __restrict__ Km, const float* __restrict__ u,
                                   float* __restrict__ part) {
    __shared__ float su[128];
    const int j  = blockIdx.x * 256 + threadIdx.x;
    const int i0 = blockIdx.y * 128;
    if (threadIdx.x < 128) su[threadIdx.x] = u[i0 + threadIdx.x];
    __syncthreads();
    float acc = 0.0f;
    #pragma unroll 8
    for (int i = 0; i < 128; ++i)
        acc += Km[(size_t)(i0 + i) * NPTS + j] * su[i];
    part[(size_t)blockIdx.y * NPTS + j] = acc;
}

// ---------------- v = b / (sum partials + eps) ----------------
__global__ void finalize_v(const float* __restrict__ part, float* __restrict__ v) {
    const int j = blockIdx.x * 256 + threadIdx.x;
    float s = 0.0f;
    #pragma unroll
    for (int c = 0; c < 32; ++c) s += part[(size_t)c * NPTS + j];
    v[j] = MARG / (s + EPSF);
}

// ---------------- loss = sum_ij u_i * K_ij * v_j * (-reg*log K_ij), two-phase ----------------
__global__ void loss_partial(const float* __restrict__ Km, const float* __restrict__ u,
                             const float* __restrict__ v, float* __restrict__ bsum) {
    __shared__ float s[256];
    const size_t total  = (size_t)NPTS * NPTS;
    const size_t stride = (size_t)LOSS_BLOCKS * 256;
    float acc = 0.0f;
    for (size_t p = (size_t)blockIdx.x * 256 + threadIdx.x; p < total; p += stride) {
        const float k = Km[p];
        if (k > 0.0f) {
            const int i = (int)(p >> 12);          // row (N = 4096 = 2^12)
            const int j = (int)(p & (NPTS - 1));   // col
            const float cost = -REGF * logf(k);    // exact inverse of K = exp(-cost/reg)
            acc += u[i] * k * v[j] * cost;
        }
    }
    s[threadIdx.x] = acc;
    __syncthreads();
    for (int st = 128; st > 0; st >>= 1) {
        if (threadIdx.x < st) s[threadIdx.x] += s[threadIdx.x + st];
        __syncthreads();
    }
    if (threadIdx.x == 0) bsum[blockIdx.x] = s[0];
}

__global__ void loss_final(const float* __restrict__ bsum, float* __restrict__ out) {
    __shared__ float s[256];
    const int t = threadIdx.x;
    float a = 0.0f;
    for (int c = t; c < LOSS_BLOCKS; c += 256) a += bsum[c];
    s[t] = a;
    __syncthreads();
    for (int st = 128; st > 0; st >>= 1) {
        if (t < st) s[t] += s[t + st];
        __syncthreads();
    }
    if (t == 0) out[0] = s[0];
}

// ---------------- host-side orchestration ----------------
extern "C" void kernel_launch(void* const* d_in, const int* in_sizes, int n_in,
                              void* d_out, int out_size, void* d_ws, size_t ws_size,
                              hipStream_t stream) {
    (void)in_sizes; (void)n_in; (void)out_size; (void)ws_size;
    const float* X = (const float*)d_in[0];   // audio_features [4096,512] f32
    const float* Y = (const float*)d_in[1];   // text_features  [4096,512] f32
    float* out = (float*)d_out;

    // Workspace layout (floats): K | Xs | Ys | u | v | partials(32*N) | bsum
    float* ws   = (float*)d_ws;
    const size_t NN = (size_t)NPTS * NPTS;
    float* Km   = ws;
    float* Xs   = ws + NN;
    float* Ys   = Xs + NPTS;
    float* u    = Ys + NPTS;
    float* v    = u + NPTS;
    float* part = v + NPTS;
    float* bsum = part + (size_t)32 * NPTS;   // ~67.7 MB total

    init_uv<<<(NPTS + 255) / 256, 256, 0, stream>>>(u, v);
    row_norms<<<NPTS, 256, 0, stream>>>(X, Y, Xs, Ys);
    gemm_expk<<<dim3(NPTS / 128, NPTS / 64), 256, 0, stream>>>(X, Y, Xs, Ys, Km);

    for (int it = 0; it < ITERS; ++it) {
        matvec_row_u<<<NPTS, 256, 0, stream>>>(Km, v, u);
        matvec_col_partial<<<dim3(NPTS / 256, 32), 256, 0, stream>>>(Km, u, part);
        finalize_v<<<NPTS / 256, 256, 0, stream>>>(part, v);
    }

    loss_partial<<<LOSS_BLOCKS, 256, 0, stream>>>(Km, u, v, bsum);
    loss_final<<<1, 256, 0, stream>>>(bsum, out);
}